// MultiNetworkRNN_59889023975875
// MI455X (gfx1250) — compile-verified
//
#include <hip/hip_runtime.h>

#define N_NET   32
#define N_UNITS 2048
#define N_EXC   1638
#define N_INH   410
#define DT_OVER_TAU 0.1f

typedef __attribute__((ext_vector_type(2))) float v2f;
typedef __attribute__((ext_vector_type(8))) float v8f;

// ---------------------------------------------------------------------------
// Kernel 1: inter[i,a] = sum_j sum_b W_inter[i,j,a,b] * r[i, N_EXC+b]
// Contraction runs along the contiguous b axis -> coalesced row-dot per wave.
// ---------------------------------------------------------------------------
#define INTER_WAVES 8
__global__ __launch_bounds__(256) void inter_kernel(const float* __restrict__ W_inter,
                                                    const float* __restrict__ r,
                                                    float* __restrict__ inter) {
    __shared__ float s_ri[N_INH];
    const int blocks_per_net = (N_INH + INTER_WAVES - 1) / INTER_WAVES; // 52
    const int net  = blockIdx.x / blocks_per_net;
    const int a    = (blockIdx.x % blocks_per_net) * INTER_WAVES + (threadIdx.x >> 5);
    const int lane = threadIdx.x & 31;

    for (int t = threadIdx.x; t < N_INH; t += blockDim.x)
        s_ri[t] = r[net * N_UNITS + N_EXC + t];
    __syncthreads();

    if (a < N_INH) {
        float acc = 0.0f;
        const float* base = W_inter + (size_t)net * N_NET * (N_INH * N_INH)
                                    + (size_t)a * N_INH;
        #pragma unroll 1
        for (int j = 0; j < N_NET; ++j) {
            const float* row = base + (size_t)j * (N_INH * N_INH);
            #pragma unroll 4
            for (int b = lane; b < N_INH; b += 32)
                acc += row[b] * s_ri[b];
        }
        #pragma unroll
        for (int off = 16; off > 0; off >>= 1)
            acc += __shfl_down(acc, off, 32);
        if (lane == 0) inter[net * N_INH + a] = acc;
    }
}

// ---------------------------------------------------------------------------
// WMMA matvec accumulation: one wave, 16 output columns.
// A (16x4 f32): all 16 rows = r[p0..p0+3]; lanes 0-15 hold K={0,1},
//               lanes 16-31 hold K={2,3} (per ISA A 16x4 layout).
// B (4x16 f32): lane n holds column n of rows K={khalf, khalf+1}.
// C (16x16 f32) accumulates across the K loop; all D rows identical.
// ---------------------------------------------------------------------------
__device__ __forceinline__ v8f wmma_step(v2f a, v2f b, v8f c) {
    return __builtin_amdgcn_wmma_f32_16x16x4_f32(
        /*neg_a=*/false, a, /*neg_b=*/false, b,
        /*c_mod=*/(short)0, c, /*reuse_a=*/false, /*reuse_b=*/false);
}

__device__ __forceinline__ v8f matvec_acc(const float* __restrict__ W, int ldq, int K,
                                          const float* __restrict__ rv,
                                          int col, int khalf, v8f c) {
    const float* wp0 = W + (size_t)khalf * ldq + col;
    const float* wp1 = wp0 + ldq;
    const size_t step = (size_t)4 * ldq;
    const int Kfull = K & ~3;
    #pragma unroll 2
    for (int p0 = 0; p0 < Kfull; p0 += 4) {
        v2f a; a.x = rv[p0 + khalf]; a.y = rv[p0 + khalf + 1];
        v2f b; b.x = *wp0;           b.y = *wp1;
        c = wmma_step(a, b, c);
        wp0 += step; wp1 += step;
    }
    if (Kfull < K) {  // tail (K % 4 == 2 for all our K): zero A, clamp B rows
        int k0 = Kfull + khalf, k1 = k0 + 1;
        v2f a; a.x = (k0 < K) ? rv[k0] : 0.0f;
               a.y = (k1 < K) ? rv[k1] : 0.0f;
        int r0 = (k0 < K) ? k0 : (K - 1);
        int r1 = (k1 < K) ? k1 : (K - 1);
        v2f b; b.x = W[(size_t)r0 * ldq + col];
               b.y = W[(size_t)r1 * ldq + col];
        c = wmma_step(a, b, c);
    }
    return c;
}

// ---------------------------------------------------------------------------
// Kernel 2: per net, WMMA matvecs + bias + relu + Euler update.
// ---------------------------------------------------------------------------
#define E_TILES 103                       // ceil(1638/16)
#define I_TILES 26                        // ceil(410/16)
#define TILES_PER_NET (E_TILES + I_TILES) // 129
#define WAVES_PER_BLOCK 4
#define BLOCKS_PER_NET ((TILES_PER_NET + WAVES_PER_BLOCK - 1) / WAVES_PER_BLOCK) // 33

__global__ __launch_bounds__(128) void rnn_kernel(
        const float* __restrict__ unit_input, const float* __restrict__ r,
        const float* __restrict__ W_ee, const float* __restrict__ W_ei,
        const float* __restrict__ W_ie, const float* __restrict__ W_ii,
        const float* __restrict__ inter, float* __restrict__ out) {
    __shared__ float s_r[N_UNITS];
    const int net   = blockIdx.x / BLOCKS_PER_NET;
    const int tile0 = (blockIdx.x % BLOCKS_PER_NET) * WAVES_PER_BLOCK;
    const float* rn = r + net * N_UNITS;

    { // stage r[net] into LDS, vectorized
        const float4* rn4 = (const float4*)rn;
        float4* s4 = (float4*)s_r;
        for (int t = threadIdx.x; t < N_UNITS / 4; t += blockDim.x)
            s4[t] = rn4[t];
    }
    __syncthreads();

    const int wave = threadIdx.x >> 5;
    const int lane = threadIdx.x & 31;
    const int tile = tile0 + wave;
    if (tile >= TILES_PER_NET) return;      // wave-uniform: EXEC stays all-ones

    const int khalf = (lane < 16) ? 0 : 2;
    const int n     = lane & 15;

    v8f c = {0.f, 0.f, 0.f, 0.f, 0.f, 0.f, 0.f, 0.f};
    float bias;
    int   q;       // output column in [0, N_UNITS)
    bool  valid;

    if (tile < E_TILES) {
        const int q0  = tile * 16;
        int col = q0 + n; if (col >= N_EXC) col = N_EXC - 1;   // clamp partial tile
        const float* Wee = W_ee + (size_t)net * N_EXC * N_EXC;
        const float* Wie = W_ie + (size_t)net * N_INH * N_EXC;
        c = matvec_acc(Wee, N_EXC, N_EXC, s_r,         col, khalf, c);
        c = matvec_acc(Wie, N_EXC, N_INH, s_r + N_EXC, col, khalf, c);
        q     = q0 + n;
        valid = (q < N_EXC) && (lane < 16);
        bias  = unit_input[net * N_UNITS + col];
    } else {
        const int q0  = (tile - E_TILES) * 16;
        int col = q0 + n; if (col >= N_INH) col = N_INH - 1;
        const float* Wei = W_ei + (size_t)net * N_EXC * N_INH;
        const float* Wii = W_ii + (size_t)net * N_INH * N_INH;
        c = matvec_acc(Wei, N_INH, N_EXC, s_r,         col, khalf, c);
        c = matvec_acc(Wii, N_INH, N_INH, s_r + N_EXC, col, khalf, c);
        q     = N_EXC + q0 + n;
        valid = (q0 + n < N_INH) && (lane < 16);
        bias  = unit_input[net * N_UNITS + N_EXC + col] + inter[net * N_INH + col];
    }

    // D row 0: VGPR0 of lanes 0-15 holds N = lane (all rows identical anyway)
    const float total = c[0] + bias;
    const float phi   = total > 0.0f ? total : 0.0f;
    const int   qs    = (q < N_UNITS) ? q : (N_UNITS - 1);
    const float rv    = s_r[qs];
    const float rnew  = rv + DT_OVER_TAU * (phi - rv);
    if (valid) out[net * N_UNITS + q] = rnew;
}

// ---------------------------------------------------------------------------
extern "C" void kernel_launch(void* const* d_in, const int* in_sizes, int n_in,
                              void* d_out, int out_size, void* d_ws, size_t ws_size,
                              hipStream_t stream) {
    (void)in_sizes; (void)n_in; (void)out_size; (void)ws_size;
    const float* unit_input = (const float*)d_in[0];
    const float* r          = (const float*)d_in[1];
    const float* W_ee       = (const float*)d_in[2];
    const float* W_ei       = (const float*)d_in[3];
    const float* W_ie       = (const float*)d_in[4];
    const float* W_ii       = (const float*)d_in[5];
    const float* W_inter    = (const float*)d_in[6];
    float* out   = (float*)d_out;
    float* inter = (float*)d_ws;   // N_NET * N_INH floats, fully overwritten

    const int g1 = N_NET * ((N_INH + INTER_WAVES - 1) / INTER_WAVES);
    inter_kernel<<<g1, 256, 0, stream>>>(W_inter, r, inter);

    const int g2 = N_NET * BLOCKS_PER_NET;
    rnn_kernel<<<g2, 128, 0, stream>>>(unit_input, r, W_ee, W_ei, W_ie, W_ii,
                                       inter, out);
}